// Position_attention_layer_19670950216607
// MI455X (gfx1250) — compile-verified
//
#include <hip/hip_runtime.h>
#include <cstdint>

// ---------------------------------------------------------------------------
// Problem constants (from the reference): x:(8,4096,1024) f32,
// kernel:(4106,1024) f32, W:(1024,) f32, b:(4106,) f32. out:(8,4096,1024) f32.
// ---------------------------------------------------------------------------
#define B_LEN 8
#define S_LEN 4096
#define F_LEN 1024
#define WIN   10
#define KROWS 4106          // S + WINDOWS
#define EPSV  1e-7f

typedef __attribute__((ext_vector_type(2))) float v2f;
typedef __attribute__((ext_vector_type(4))) float v4f;
typedef __attribute__((ext_vector_type(8))) float v8f;

// ---------------------------------------------------------------------------
// Kernel 1: a[row] = exp(tanh(dot(kernel[row,:], W) + b[row])), row in [0,4106)
// One wave32 per row; vector loads; cross-lane shuffle reduction.
// ---------------------------------------------------------------------------
__global__ __launch_bounds__(256)
void gemv_a_kernel(const float* __restrict__ kern,
                   const float* __restrict__ Wv,
                   const float* __restrict__ bvec,
                   float* __restrict__ a_out)
{
    const int row  = blockIdx.x * 8 + (threadIdx.x >> 5);
    const int lane = threadIdx.x & 31;
    if (row >= KROWS) return;

    const v4f* kr = reinterpret_cast<const v4f*>(kern + (size_t)row * F_LEN);
    const v4f* w4 = reinterpret_cast<const v4f*>(Wv);

    float sum = 0.0f;
#pragma unroll
    for (int i = lane; i < F_LEN / 4; i += 32) {
        v4f k4 = kr[i];
        v4f v4 = w4[i];
        sum += k4.x * v4.x + k4.y * v4.y + k4.z * v4.z + k4.w * v4.w;
    }
#pragma unroll
    for (int off = 16; off > 0; off >>= 1)
        sum += __shfl_down(sum, off, 32);

    if (lane == 0)
        a_out[row] = expf(tanhf(sum + bvec[row]));
}

// ---------------------------------------------------------------------------
// Kernel 2: invden[s] = 1 / (sum_{w<10} a[s+w] + eps), s in [0,4096)
// ---------------------------------------------------------------------------
__global__ __launch_bounds__(256)
void invden_kernel(const float* __restrict__ a, float* __restrict__ invden)
{
    const int s = blockIdx.x * 256 + threadIdx.x;
    if (s >= S_LEN) return;
    float t = 0.0f;
#pragma unroll
    for (int w = 0; w < WIN; ++w) t += a[s + w];
    invden[s] = 1.0f / (t + EPSV);
}

// ---------------------------------------------------------------------------
// Kernel 3 (fused, dominant): per 16(s)x16(f) tile,
//   alltemp_tile = A_band(16x28) x kernel_rows(28x16)   via V_WMMA_F32_16X16X4_F32
//   tile = tanh(tile)  -> LDS
//   out[b,s,f] = x[b,s,f] + tile   for all 8 batches, v4f + non-temporal.
//
// Block = 256 threads = 8 waves. blockIdx.x = s-tile (256), blockIdx.y =
// f-group of 128 columns (8). Wave w owns f-tile = blockIdx.y*8 + w.
//
// WMMA f32 16x16x4 VGPR layouts (ISA 7.12.2):
//   A (16x4): lanes 0-15 -> M=lane, K={0,1}; lanes 16-31 -> M=lane-16, K={2,3}
//   B (4x16): lanes 0-15 -> N=lane, K={0,1}; lanes 16-31 -> N=lane-16, K={2,3}
//   C/D: vgpr r holds M = r + 8*(lane>>4), N = lane&15
// ---------------------------------------------------------------------------
__global__ __launch_bounds__(256)
void attn_add_kernel(const float* __restrict__ x,
                     const float* __restrict__ kern,
                     const float* __restrict__ a,
                     const float* __restrict__ invden,
                     float* __restrict__ out)
{
    __shared__ float tile[16][132];   // 132-stride: rows 8 apart land 32 banks apart

    const int st   = blockIdx.x;            // s-tile index, 0..255
    const int fg   = blockIdx.y;            // f-group,      0..7
    const int wave = threadIdx.x >> 5;      // 0..7
    const int lane = threadIdx.x & 31;
    const int g    = lane >> 4;             // half-wave: 0 or 1
    const int m    = lane & 15;             // A row / B col within tile

    const int s0 = st * 16;
    const int f0 = fg * 128 + wave * 16;

    const float inv = invden[s0 + m];       // per-lane row denominator

    // ---- Phase 1a: build all 7 A fragments, branch-free (selects, no EXEC
    // divergence). a[] index is clamped so every load is in-bounds; the band
    // predicate (m <= k < m+10) zeroes padding / clamped entries.
    v2f Af[7];
#pragma unroll
    for (int c = 0; c < 7; ++c) {
        const int k0 = 4 * c + 2 * g;
        const int k1 = k0 + 1;
        const int kmax = KROWS - 1 - s0;    // largest in-bounds k offset
        const int i0 = s0 + ((k0 < kmax) ? k0 : kmax);
        const int i1 = s0 + ((k1 < kmax) ? k1 : kmax);
        const float av0 = a[i0];
        const float av1 = a[i1];
        Af[c].x = (k0 >= m && k0 < m + WIN) ? av0 * inv : 0.0f;
        Af[c].y = (k1 >= m && k1 < m + WIN) ? av1 * inv : 0.0f;
    }

    // ---- Phase 1b: issue all 14 B loads up front so they pipeline.
    v2f Bf[7];
#pragma unroll
    for (int c = 0; c < 7; ++c) {
        const int k0 = 4 * c + 2 * g;
        const int k1 = k0 + 1;
        int r0 = s0 + k0; if (r0 > KROWS - 1) r0 = KROWS - 1;   // A there is 0
        int r1 = s0 + k1; if (r1 > KROWS - 1) r1 = KROWS - 1;
        Bf[c].x = kern[(size_t)r0 * F_LEN + f0 + m];
        Bf[c].y = kern[(size_t)r1 * F_LEN + f0 + m];
    }

    // ---- Phase 1c: chained WMMAs (accumulator RAW is hazard-free).
    v8f acc = {};
#pragma unroll
    for (int c = 0; c < 7; ++c) {
        acc = __builtin_amdgcn_wmma_f32_16x16x4_f32(
                  /*neg_a=*/false, Af[c], /*neg_b=*/false, Bf[c],
                  /*c_mod=*/(short)0, acc,
                  /*reuse_a=*/false, /*reuse_b=*/false);
    }

    // tanh + stage in LDS. vgpr r -> row (r + 8g), col (wave*16 + m)
#pragma unroll
    for (int r = 0; r < 8; ++r)
        tile[r + 8 * g][wave * 16 + m] = tanhf(acc[r]);

    __syncthreads();

    // ---- Phase 2: broadcast add over all 8 batches (dominant HBM traffic,
    // ~268 MB of ~285 MB). Each byte touched exactly once -> non-temporal.
    const v4f* x4 = reinterpret_cast<const v4f*>(x);
    v4f*       o4 = reinterpret_cast<v4f*>(out);
    const int fbase4 = fg * 32;            // f-group start in float4 units

#pragma unroll
    for (int j = 0; j < 16; ++j) {
        const int i  = threadIdx.x + 256 * j;
        const int f4 = i & 31;
        const int sr = (i >> 5) & 15;
        const int bb = i >> 9;

        v4f t = *reinterpret_cast<const v4f*>(&tile[sr][f4 * 4]);

        const size_t gidx = ((size_t)bb * S_LEN + (s0 + sr)) * (F_LEN / 4)
                          + fbase4 + f4;
        v4f xv = __builtin_nontemporal_load(&x4[gidx]);
        v4f ov = xv + t;
        __builtin_nontemporal_store(ov, &o4[gidx]);
    }
}

// ---------------------------------------------------------------------------
// Host-side launcher. d_in order: x, kernel, W, b.
// Workspace: a[] (4106 f32) at ws+0, invden[] (4096 f32) at ws+4224 floats.
// ---------------------------------------------------------------------------
extern "C" void kernel_launch(void* const* d_in, const int* in_sizes, int n_in,
                              void* d_out, int out_size, void* d_ws, size_t ws_size,
                              hipStream_t stream)
{
    const float* x    = (const float*)d_in[0];
    const float* kern = (const float*)d_in[1];
    const float* Wv   = (const float*)d_in[2];
    const float* bvec = (const float*)d_in[3];
    float*       out  = (float*)d_out;

    float* a_buf   = (float*)d_ws;          // 4106 floats
    float* inv_buf = a_buf + 4224;          // 4096 floats (aligned past a_buf)

    // 1) a = exp(tanh(kernel @ W + b)); one wave per row
    gemv_a_kernel<<<(KROWS + 7) / 8, 256, 0, stream>>>(kern, Wv, bvec, a_buf);

    // 2) window denominators
    invden_kernel<<<(S_LEN + 255) / 256, 256, 0, stream>>>(a_buf, inv_buf);

    // 3) fused WMMA banded matmul + tanh + batch-broadcast add
    dim3 grid(S_LEN / 16, F_LEN / 128, 1);
    attn_add_kernel<<<grid, 256, 0, stream>>>(x, kern, a_buf, inv_buf, out);
}